// NMRGNN_9208409882677
// MI455X (gfx1250) — compile-verified
//
#include <hip/hip_runtime.h>
#include <hip/hip_bf16.h>

// CDNA5 / gfx1250. fp32 end-to-end (matches reference numerics) using
// V_WMMA_F32_16X16X4_F32 for all GEMMs. Workload is L2-bandwidth bound
// (whole working set < 192 MB L2), so fp32 matrix math costs nothing.

typedef float v2f __attribute__((ext_vector_type(2)));
typedef float v8f __attribute__((ext_vector_type(8)));

#define N_NODES 100000
#define N_EDGES 1250000
#define F_IN    128
#define EQ_DIM  8
#define DIM0    136          // F_IN + EQ_DIM
#define HDIM    64
#define BN_EPS  1e-5f

#define STU 140              // LDS stride for U tile (140 % 64 = 12 -> conflict-free col reads)
#define STT 68               // LDS stride for T tile (68 % 64 = 4  -> conflict-free col reads)

// ---------------- h0 = concat(x, eq_table[eq_labels]) ----------------
__global__ void build_h0_kernel(const float* __restrict__ x,
                                const int* __restrict__ eq_labels,
                                const float* __restrict__ eq_table,
                                float* __restrict__ h) {
    int idx = blockIdx.x * blockDim.x + threadIdx.x;
    if (idx >= N_NODES * DIM0) return;
    int i = idx / DIM0;
    int f = idx - i * DIM0;
    float v;
    if (f < F_IN) v = x[i * F_IN + f];
    else          v = eq_table[eq_labels[i] * EQ_DIM + (f - F_IN)];
    h[idx] = v;
}

// ---------------- agg[dst] += h[src] over edges (float4 chunks) ------
__global__ void scatter_add_kernel(const float* __restrict__ h,
                                   const int* __restrict__ ei,  // [2, E]
                                   float* __restrict__ agg,
                                   int dim, int chunks, int total) {
    int idx = blockIdx.x * blockDim.x + threadIdx.x;
    if (idx >= total) return;
    int e = idx / chunks;
    int q = idx - e * chunks;
    int src = ei[e];
    int dst = ei[N_EDGES + e];
    const float4 v = *reinterpret_cast<const float4*>(h + src * dim + q * 4);
    float* a = agg + dst * dim + q * 4;
    atomicAdd(a + 0, v.x);
    atomicAdd(a + 1, v.y);
    atomicAdd(a + 2, v.z);
    atomicAdd(a + 3, v.w);
}

// ---------------- GIN combine + 2-layer MLP via f32 WMMA -------------
// One wave per 16-row tile; 2 waves (2 tiles) per 64-thread block.
// N = 6250 * 16 exactly -> every tile full, EXEC all-1s for WMMA.
template <int DIM>
__global__ __launch_bounds__(64)
void gin_mlp_wmma_kernel(const float* __restrict__ h,    // [N, DIM]
                         const float* __restrict__ agg,  // [N, DIM]
                         const float* __restrict__ W1,   // [DIM, 64] row-major
                         const float* __restrict__ b1,   // [64]
                         const float* __restrict__ W2,   // [64, 64]
                         const float* __restrict__ b2,   // [64]
                         const float* __restrict__ epsp, // [1]
                         float* __restrict__ out,        // [N, 64] pre-BN
                         float* __restrict__ stats) {    // [128] = {sum[64], sumsq[64]}
    __shared__ float sU[2][16 * STU];
    __shared__ float sT[2][16 * STT];

    const int lane = threadIdx.x & 31;
    const int w    = threadIdx.x >> 5;
    const int row0 = (blockIdx.x * 2 + w) * 16;
    const float se = 1.0f + epsp[0];

    float* su = sU[w];
    float* st = sT[w];

    // Stage U = (1+eps)*h + agg (coalesced; 16*DIM is a multiple of 32)
    for (int t = lane; t < 16 * DIM; t += 32) {
        int r = t / DIM;
        int c = t - r * DIM;
        int g = (row0 + r) * DIM + c;
        su[r * STU + c] = fmaf(se, h[g], agg[g]);
    }
    asm volatile("s_wait_dscnt 0" ::: "memory");  // cross-lane LDS handoff

    const int m     = lane & 15;          // col within 16-wide tile / row of A
    const int khalf = (lane >> 4) << 1;   // lanes 0-15 -> K{0,1}; 16-31 -> K{2,3}
    const int mrow  = (lane >> 4) << 3;   // C/D: lanes 16-31 hold M = r + 8

    // ---- GEMM1: T = relu(U @ W1 + b1), K = DIM ----
    v8f acc[4];
#pragma unroll
    for (int n = 0; n < 4; ++n) acc[n] = (v8f){};
#pragma unroll
    for (int kk = 0; kk < DIM; kk += 4) {
        v2f a;
        a.x = su[m * STU + kk + khalf + 0];
        a.y = su[m * STU + kk + khalf + 1];
        const float* w1r = W1 + (kk + khalf) * HDIM;
#pragma unroll
        for (int n = 0; n < 4; ++n) {
            v2f b;
            b.x = w1r[n * 16 + m];
            b.y = w1r[HDIM + n * 16 + m];
            acc[n] = __builtin_amdgcn_wmma_f32_16x16x4_f32(
                false, a, false, b, (short)0, acc[n], false, false);
        }
    }

    // bias + relu -> stage T in LDS (re-fragment as A for GEMM2)
#pragma unroll
    for (int n = 0; n < 4; ++n) {
        float bn = b1[n * 16 + m];
#pragma unroll
        for (int r = 0; r < 8; ++r) {
            float v = fmaxf(acc[n][r] + bn, 0.0f);
            st[(r + mrow) * STT + n * 16 + m] = v;
        }
    }
    asm volatile("s_wait_dscnt 0" ::: "memory");

    // ---- GEMM2: V = relu(T @ W2 + b2), K = 64 ----
#pragma unroll
    for (int n = 0; n < 4; ++n) acc[n] = (v8f){};
#pragma unroll
    for (int kk = 0; kk < HDIM; kk += 4) {
        v2f a;
        a.x = st[m * STT + kk + khalf + 0];
        a.y = st[m * STT + kk + khalf + 1];
        const float* w2r = W2 + (kk + khalf) * HDIM;
#pragma unroll
        for (int n = 0; n < 4; ++n) {
            v2f b;
            b.x = w2r[n * 16 + m];
            b.y = w2r[HDIM + n * 16 + m];
            acc[n] = __builtin_amdgcn_wmma_f32_16x16x4_f32(
                false, a, false, b, (short)0, acc[n], false, false);
        }
    }

    // bias + relu -> store pre-BN activations + BN statistics
#pragma unroll
    for (int n = 0; n < 4; ++n) {
        float bn = b2[n * 16 + m];
        float s = 0.0f, sq = 0.0f;
#pragma unroll
        for (int r = 0; r < 8; ++r) {
            float v = fmaxf(acc[n][r] + bn, 0.0f);
            out[(row0 + mrow + r) * HDIM + n * 16 + m] = v;
            s += v;
            sq = fmaf(v, v, sq);
        }
        atomicAdd(&stats[n * 16 + m], s);
        atomicAdd(&stats[64 + n * 16 + m], sq);
    }
}

// ---------------- BatchNorm (batch stats, biased var) + ReLU ---------
__global__ void bn_relu_kernel(const float* __restrict__ t2,
                               const float* __restrict__ stats,
                               const float* __restrict__ gamma,
                               const float* __restrict__ beta,
                               float* __restrict__ h) {
    int idx = blockIdx.x * blockDim.x + threadIdx.x;
    if (idx >= N_NODES * HDIM) return;
    int c = idx & (HDIM - 1);
    const float invN = 1.0f / (float)N_NODES;
    float mean = stats[c] * invN;
    float var  = stats[64 + c] * invN - mean * mean;
    float inv  = rsqrtf(var + BN_EPS);
    float v = (t2[idx] - mean) * inv * gamma[c] + beta[c];
    h[idx] = fmaxf(v, 0.0f);
}

// ---------------- out = h @ W_out + b_out ----------------------------
__global__ void out_proj_kernel(const float* __restrict__ h,
                                const float* __restrict__ Wout,
                                const float* __restrict__ bout,
                                float* __restrict__ out) {
    __shared__ float sw[HDIM];
    if (threadIdx.x < HDIM) sw[threadIdx.x] = Wout[threadIdx.x];
    __syncthreads();
    int i = blockIdx.x * blockDim.x + threadIdx.x;
    if (i >= N_NODES) return;
    const float4* hp = reinterpret_cast<const float4*>(h + i * HDIM);
    float acc = bout[0];
#pragma unroll
    for (int q = 0; q < HDIM / 4; ++q) {
        float4 v = hp[q];
        acc = fmaf(v.x, sw[4 * q + 0], acc);
        acc = fmaf(v.y, sw[4 * q + 1], acc);
        acc = fmaf(v.z, sw[4 * q + 2], acc);
        acc = fmaf(v.w, sw[4 * q + 3], acc);
    }
    out[i] = acc;
}

// Input flattening assumption: top-level setup_inputs() dict in insertion
// order (x, edge_index, eq_labels, batch), then params pytree-flattened with
// sorted dict keys: W_out, b_out, eq_table, then layers[0..2] each as
// (W1, W2, b1, b2, beta, eps, gamma).
extern "C" void kernel_launch(void* const* d_in, const int* in_sizes, int n_in,
                              void* d_out, int out_size, void* d_ws, size_t ws_size,
                              hipStream_t stream) {
    const float* x         = (const float*)d_in[0];   // [N, 128]
    const int*   ei        = (const int*)d_in[1];     // [2, E]
    const int*   eq_labels = (const int*)d_in[2];     // [N]
    /* d_in[3] = batch (unused by forward) */
    const float* W_out     = (const float*)d_in[4];   // [64]
    const float* b_out     = (const float*)d_in[5];   // [1]
    const float* eq_table  = (const float*)d_in[6];   // [64, 8]

    float* h     = (float*)d_ws;                         // [N, 136] (reused as [N,64])
    float* agg   = h   + (size_t)N_NODES * DIM0;         // [N, 136]
    float* t2    = agg + (size_t)N_NODES * DIM0;         // [N, 64]
    float* stats = t2  + (size_t)N_NODES * HDIM;         // [128]

    build_h0_kernel<<<(N_NODES * DIM0 + 255) / 256, 256, 0, stream>>>(
        x, eq_labels, eq_table, h);

    for (int l = 0; l < 3; ++l) {
        const int dim = (l == 0) ? DIM0 : HDIM;
        const int base = 7 + 7 * l;
        const float* W1    = (const float*)d_in[base + 0];
        const float* W2    = (const float*)d_in[base + 1];
        const float* b1    = (const float*)d_in[base + 2];
        const float* b2    = (const float*)d_in[base + 3];
        const float* beta  = (const float*)d_in[base + 4];
        const float* epsp  = (const float*)d_in[base + 5];
        const float* gamma = (const float*)d_in[base + 6];

        hipMemsetAsync(agg,   0, (size_t)N_NODES * dim * sizeof(float), stream);
        hipMemsetAsync(stats, 0, 128 * sizeof(float), stream);

        const int chunks = dim / 4;
        const int total  = N_EDGES * chunks;
        scatter_add_kernel<<<(total + 255) / 256, 256, 0, stream>>>(
            h, ei, agg, dim, chunks, total);

        if (l == 0) {
            gin_mlp_wmma_kernel<DIM0><<<N_NODES / 32, 64, 0, stream>>>(
                h, agg, W1, b1, W2, b2, epsp, t2, stats);
        } else {
            gin_mlp_wmma_kernel<HDIM><<<N_NODES / 32, 64, 0, stream>>>(
                h, agg, W1, b1, W2, b2, epsp, t2, stats);
        }

        bn_relu_kernel<<<(N_NODES * HDIM + 255) / 256, 256, 0, stream>>>(
            t2, stats, gamma, beta, h);
    }

    out_proj_kernel<<<(N_NODES + 255) / 256, 256, 0, stream>>>(
        h, W_out, b_out, (float*)d_out);
}